// nomem_update_27092653703301
// MI455X (gfx1250) — compile-verified
//
#include <hip/hip_runtime.h>
#include <stdint.h>

#define HI_BINS   65536
#define TPB_HIST  1024
#define HIST_BLOCKS 256
#define TILE_ELEMS (TPB_HIST * 4)   // 4096 f32 per tile: 16B per lane (b128)

typedef float vf4 __attribute__((ext_vector_type(4)));   // clang-native 128-bit vector

// ---- order-preserving key transform (monotonic with float compare; -0 == +0) ----
__device__ __forceinline__ unsigned topk_key_of(float f) {
  unsigned u = __float_as_uint(f);
  if (u == 0x80000000u) u = 0u;                       // canonicalize -0 -> +0
  return (u & 0x80000000u) ? ~u : (u | 0x80000000u);
}
__device__ __forceinline__ float topk_key_to_float(unsigned k) {
  unsigned u = (k & 0x80000000u) ? (k & 0x7fffffffu) : ~k;
  return __uint_as_float(u);
}

// ---- histogram kernel with async global->LDS double-buffered staging ----
// mode 0: histogram of top 16 key bits
// mode 1: histogram of low 16 key bits restricted to sel[0]          (fallback)
// mode 2: mode 1 + inline decided-mask write + candidate-index list  (fused)
__global__ __launch_bounds__(TPB_HIST)
void topk_hist_kernel(const float* __restrict__ x, unsigned long long n,
                      unsigned* __restrict__ ghist,
                      const unsigned* __restrict__ sel, int mode,
                      vf4* __restrict__ out4, float* __restrict__ out,
                      unsigned* __restrict__ cand, unsigned* __restrict__ cand_count,
                      unsigned cap) {
  __shared__ unsigned hist[HI_BINS];          // 256 KB
  __shared__ float stage[2][TILE_ELEMS];      // 32 KB double buffer

  for (int i = threadIdx.x; i < HI_BINS; i += TPB_HIST) hist[i] = 0u;
  __syncthreads();

  const unsigned bin_hi = (mode != 0) ? sel[0] : 0u;

  const unsigned long long tiles = n / (unsigned long long)TILE_ELEMS;
  const unsigned lds0 = (unsigned)(uintptr_t)(&stage[0][threadIdx.x * 4]);
  const unsigned lds1 = (unsigned)(uintptr_t)(&stage[1][threadIdx.x * 4]);

  unsigned long long t = blockIdx.x;
  int buf = 0;
  if (t < tiles) {
    unsigned long long g = (unsigned long long)(uintptr_t)
        (x + t * TILE_ELEMS + (unsigned long long)threadIdx.x * 4ull);
    asm volatile("global_load_async_to_lds_b128 %0, %1, off"
                 :: "v"(lds0), "v"(g) : "memory");
  }
  while (t < tiles) {
    const unsigned long long tn = t + gridDim.x;
    if (tn < tiles) {                          // prefetch next tile (uniform branch)
      const unsigned ldsn = buf ? lds0 : lds1;
      unsigned long long g = (unsigned long long)(uintptr_t)
          (x + tn * TILE_ELEMS + (unsigned long long)threadIdx.x * 4ull);
      asm volatile("global_load_async_to_lds_b128 %0, %1, off"
                   :: "v"(ldsn), "v"(g) : "memory");
      asm volatile("s_wait_asynccnt 0x1" ::: "memory");   // oldest (current) done
    } else {
      asm volatile("s_wait_asynccnt 0x0" ::: "memory");
    }
    const float* sp = &stage[buf][threadIdx.x * 4];       // wave-private slice
    if (mode == 0) {
#pragma unroll
      for (int e = 0; e < 4; ++e)
        atomicAdd(&hist[topk_key_of(sp[e]) >> 16], 1u);
    } else if (mode == 1) {
#pragma unroll
      for (int e = 0; e < 4; ++e) {
        const unsigned key = topk_key_of(sp[e]);
        if ((key >> 16) == bin_hi) atomicAdd(&hist[key & 0xffffu], 1u);
      }
    } else {                                   // mode 2: fused hist-lo + mask + cands
      float m[4];
#pragma unroll
      for (int e = 0; e < 4; ++e) {
        const unsigned key = topk_key_of(sp[e]);
        const unsigned hi  = key >> 16;
        m[e] = (hi > bin_hi) ? 1.0f : 0.0f;    // ties stay 0 until fixup
        if (hi == bin_hi) {
          atomicAdd(&hist[key & 0xffffu], 1u);
          const unsigned slot = atomicAdd(cand_count, 1u);
          if (slot < cap)
            cand[slot] = (unsigned)(t * TILE_ELEMS + threadIdx.x * 4u + e);
        }
      }
      vf4 mv; mv.x = m[0]; mv.y = m[1]; mv.z = m[2]; mv.w = m[3];
      __builtin_nontemporal_store(mv, &out4[t * TPB_HIST + threadIdx.x]);
    }
    t = tn;
    buf ^= 1;
  }

  // scalar tail (n % TILE_ELEMS), block 0 only, direct loads
  if (blockIdx.x == 0) {
    for (unsigned long long i = tiles * TILE_ELEMS + threadIdx.x; i < n; i += TPB_HIST) {
      const unsigned key = topk_key_of(x[i]);
      const unsigned hi  = key >> 16;
      if (mode == 0) atomicAdd(&hist[hi], 1u);
      else {
        if (hi == bin_hi) {
          atomicAdd(&hist[key & 0xffffu], 1u);
          if (mode == 2) {
            const unsigned slot = atomicAdd(cand_count, 1u);
            if (slot < cap) cand[slot] = (unsigned)i;
          }
        }
        if (mode == 2) out[i] = (hi > bin_hi) ? 1.0f : 0.0f;
      }
    }
  }

  __syncthreads();
  for (int i = threadIdx.x; i < HI_BINS; i += TPB_HIST) {
    const unsigned c = hist[i];
    if (c) atomicAdd(&ghist[i], c);            // zero-skip: ~9K live bins for normal data
  }
}

// ---- single-block scan of 65536 bins -> bin containing `rank` ----
__global__ __launch_bounds__(1024)
void topk_scan_kernel(const unsigned* __restrict__ ghist, unsigned* sel,
                      unsigned rank_imm, int pass) {
  __shared__ unsigned scanb[1024];
  const int t = threadIdx.x;
  const unsigned rank = (pass == 0) ? rank_imm : sel[1];
  const unsigned base = (unsigned)t * 64u;

  unsigned s = 0;
  for (int j = 0; j < 64; ++j) s += ghist[base + j];
  scanb[t] = s;
  __syncthreads();
  for (int off = 1; off < 1024; off <<= 1) {   // Hillis-Steele inclusive scan
    const unsigned add = (t >= off) ? scanb[t - off] : 0u;
    __syncthreads();
    scanb[t] += add;
    __syncthreads();
  }
  const unsigned incl = scanb[t];
  const unsigned excl = incl - s;
  if (rank >= excl && rank < incl) {           // exactly one winner thread
    unsigned acc = excl, bin = base, r2 = 0;
    for (int j = 0; j < 64; ++j) {
      const unsigned c = ghist[base + j];
      if (rank < acc + c) { bin = base + j; r2 = rank - acc; break; }
      acc += c;
    }
    if (pass == 0) { sel[0] = bin; sel[1] = r2; }
    else {
      const unsigned thrkey = (sel[0] << 16) | bin;
      sel[2] = thrkey;
      ((float*)sel)[3] = topk_key_to_float(thrkey);
    }
  }
}

// ---- fused-path fixup: resolve the tie candidates against the exact threshold ----
__global__ __launch_bounds__(256)
void topk_fixup_kernel(const float* __restrict__ x, float* __restrict__ out,
                       const unsigned* __restrict__ sel,
                       const unsigned* __restrict__ cand,
                       const unsigned* __restrict__ cand_count, unsigned cap) {
  const float thr = ((const float*)sel)[3];
  unsigned m = *cand_count;
  if (m > cap) m = cap;
  const unsigned stride = gridDim.x * blockDim.x;
  for (unsigned i = blockIdx.x * blockDim.x + threadIdx.x; i < m; i += stride) {
    const unsigned idx = cand[i];
    out[idx] = (x[idx] >= thr) ? 1.0f : 0.0f;
  }
}

// ---- fallback full mask pass: mask = (x >= thr), b128 in / b128 NT out ----
__global__ __launch_bounds__(256)
void topk_mask_kernel(const vf4* __restrict__ x4, vf4* __restrict__ o4,
                      const float* __restrict__ x, float* __restrict__ o,
                      unsigned long long n4, unsigned long long n,
                      const unsigned* __restrict__ sel) {
  const float thr = ((const float*)sel)[3];
  const unsigned long long stride = (unsigned long long)gridDim.x * blockDim.x;
  unsigned long long i = (unsigned long long)blockIdx.x * blockDim.x + threadIdx.x;
  for (unsigned long long j = i; j < n4; j += stride) {
    const vf4 v = x4[j];
    vf4 m;
    m.x = (v.x >= thr) ? 1.0f : 0.0f;
    m.y = (v.y >= thr) ? 1.0f : 0.0f;
    m.z = (v.z >= thr) ? 1.0f : 0.0f;
    m.w = (v.w >= thr) ? 1.0f : 0.0f;
    __builtin_nontemporal_store(m, &o4[j]);
  }
  const unsigned long long tail0 = n4 * 4ull;
  if (i < n - tail0) {
    const unsigned long long idx = tail0 + i;
    o[idx] = (x[idx] >= thr) ? 1.0f : 0.0f;
  }
}

__global__ __launch_bounds__(256)
void topk_zero_kernel(float* __restrict__ o, unsigned long long n) {
  const unsigned long long stride = (unsigned long long)gridDim.x * blockDim.x;
  for (unsigned long long i = (unsigned long long)blockIdx.x * blockDim.x + threadIdx.x;
       i < n; i += stride)
    o[i] = 0.0f;
}

extern "C" void kernel_launch(void* const* d_in, const int* in_sizes, int n_in,
                              void* d_out, int out_size, void* d_ws, size_t ws_size,
                              hipStream_t stream) {
  const float* x = (const float*)d_in[0];
  float* out = (float*)d_out;
  const unsigned long long n = (unsigned long long)in_sizes[0];

  long long k = (long long)((double)n * 0.9);   // same trunc semantics as Python int()
  if (k <= 0) {
    topk_zero_kernel<<<2048, 256, 0, stream>>>(out, n);
    return;
  }
  if (k > (long long)n) k = (long long)n;
  const unsigned rank = (unsigned)(n - (unsigned long long)k); // ascending 0-based rank

  // workspace layout: [hist_hi 256K][hist_lo 256K][sel 8 u32][cand_count .. 16][cand ...]
  unsigned* ghist_hi   = (unsigned*)d_ws;
  unsigned* ghist_lo   = ghist_hi + HI_BINS;
  unsigned* sel        = ghist_lo + HI_BINS;    // [bin_hi, r2, thrkey, thr_bits]
  unsigned* cand_count = sel + 8;
  unsigned* cand       = sel + 16;
  const size_t fixed_bytes = (size_t)(2 * HI_BINS + 16) * sizeof(unsigned);
  const unsigned cap = (ws_size > fixed_bytes)
                     ? (unsigned)((ws_size - fixed_bytes) / sizeof(unsigned)) : 0u;
  // Fused path needs candidate room: require >=1M slots (~10x the ~88K ties expected
  // for N(0,1) data at 64M elements) and 32-bit indexability.
  const bool fused = (cap >= (1u << 20)) && (n < 0xFFFFFFFFull);

  (void)hipMemsetAsync(d_ws, 0, fixed_bytes, stream);

  const unsigned long long n4 = n / 4ull;
  topk_hist_kernel<<<HIST_BLOCKS, TPB_HIST, 0, stream>>>(
      x, n, ghist_hi, sel, 0, (vf4*)out, out, cand, cand_count, cap);
  topk_scan_kernel<<<1, 1024, 0, stream>>>(ghist_hi, sel, rank, 0);
  topk_hist_kernel<<<HIST_BLOCKS, TPB_HIST, 0, stream>>>(
      x, n, ghist_lo, sel, fused ? 2 : 1, (vf4*)out, out, cand, cand_count, cap);
  topk_scan_kernel<<<1, 1024, 0, stream>>>(ghist_lo, sel, 0u, 1);
  if (fused) {
    topk_fixup_kernel<<<256, 256, 0, stream>>>(x, out, sel, cand, cand_count, cap);
  } else {
    topk_mask_kernel<<<2048, 256, 0, stream>>>((const vf4*)x, (vf4*)out,
                                               x, out, n4, n, sel);
  }
}